// LearnableDemosaick_52888227283268
// MI455X (gfx1250) — compile-verified
//
#include <hip/hip_runtime.h>
#include <hip/hip_bf16.h>

// Fused learnable-demosaick for MI455X (gfx1250, wave32).
// One pass: async-stage haloed tile to LDS (ASYNCcnt path) -> green (4x 9-tap convs)
// -> closed-form 4-phase bilinear chroma from a register-cached 3x6 window
// -> non-temporal b128 stores of R,G,B planes (keep L2 for input halo reuse).

namespace {

constexpr int WT  = 128;        // output tile width
constexpr int HT  = 32;         // output tile height
constexpr int INW = WT + 10;    // 138: input tile width  (halo 5 = conv 4 + chroma 1)
constexpr int INH = HT + 10;    // 42 : input tile height
constexpr int GTW = WT + 2;     // 130: green/diff tile width  (halo 1)
constexpr int GTH = HT + 2;     // 34
constexpr int NT  = 256;        // 8 wave32s per block

typedef float v4f __attribute__((ext_vector_type(4)));

// CDNA5 async global->LDS copy. VDST VGPR holds the wave-relative LDS byte
// offset = low 32 bits of the generic (flat) shared-memory address.
__device__ __forceinline__ void async_copy_f32(const float* gsrc, const float* ldst) {
  unsigned lds_off = (unsigned)(unsigned long long)ldst;
  asm volatile("global_load_async_to_lds_b32 %0, %1, off"
               :: "v"(lds_off), "v"(gsrc)
               : "memory");
}

__device__ __forceinline__ void wait_async0() {
  asm volatile("s_wait_asynccnt 0" ::: "memory");
}

__global__ __launch_bounds__(NT)
void demosaic_fused_gfx1250(const float* __restrict__ mosaick,
                            const float* __restrict__ gfilt,
                            const float* __restrict__ grad_filt,
                            float* __restrict__ out,
                            int H, int W) {
  __shared__ float s_in[INH * INW];   // staged mosaick tile (edge-replicated)
  __shared__ float s_g [GTH * GTW];   // green
  __shared__ float s_d [GTH * GTW];   // (mosaick - green), zero outside image

  const int tid = threadIdx.x;
  const int gx0 = blockIdx.x * WT;
  const int gy0 = blockIdx.y * HT;
  const int b   = blockIdx.z;
  const size_t plane = (size_t)H * (size_t)W;
  const float* mp = mosaick + (size_t)b * plane;

  // Filters -> registers (uniform loads).
  float gf[9], df[9];
#pragma unroll
  for (int r = 0; r < 9; ++r) { gf[r] = gfilt[r]; df[r] = grad_filt[r]; }

  // ---- Phase 1: async-stage clamped input tile (edge-replicate padding for free)
  for (int j = tid; j < INH * INW; j += NT) {
    int row = j / INW;
    int col = j - row * INW;
    int gy = gy0 - 5 + row;
    int gx = gx0 - 5 + col;
    gy = gy < 0 ? 0 : (gy > H - 1 ? H - 1 : gy);
    gx = gx < 0 ? 0 : (gx > W - 1 ? W - 1 : gx);
    async_copy_f32(mp + (size_t)gy * W + gx, &s_in[j]);
  }
  wait_async0();
  __syncthreads();

  // ---- Phase 2: green + masked chroma diff on (HT+2)x(WT+2)
  for (int j = tid; j < GTH * GTW; j += NT) {
    int ly = j / GTW;
    int lx = j - ly * GTW;
    const float* rowp = &s_in[(ly + 4) * INW + lx];        // horizontal taps
    const float* colp = &s_in[ ly      * INW + (lx + 4)];  // vertical taps
    float ih = 0.f, dx = 0.f, iv = 0.f, dyv = 0.f;
#pragma unroll
    for (int r = 0; r < 9; ++r) {
      float mh = rowp[r];
      float mv = colp[r * INW];
      ih  = fmaf(mh, gf[r], ih);
      dx  = fmaf(mh, df[r], dx);
      iv  = fmaf(mv, gf[r], iv);
      dyv = fmaf(mv, df[r], dyv);
    }
    float m0  = rowp[4];                    // mosaick at this pixel
    float adx = fabsf(dx), ady = fabsf(dyv);
    float wh  = ady * __builtin_amdgcn_rcpf(adx + ady + 1e-5f);
    float gi  = wh * ih + (1.f - wh) * iv;
    int gy = gy0 - 1 + ly;
    int gx = gx0 - 1 + lx;
    float green = (((gx + gy) & 1) == 0) ? m0 : gi;        // G sites keep mosaick
    bool in_img = (gx >= 0) && (gx < W) && (gy >= 0) && (gy < H);
    s_g[j] = green;
    s_d[j] = in_img ? (m0 - green) : 0.f;   // zero-pad for the 3x3 chroma conv
  }
  __syncthreads();

  // ---- Phase 3: closed-form 4-phase bilinear chroma from register-cached window
  float* rp = out + ((size_t)b * 3 + 0) * plane;
  float* gp = out + ((size_t)b * 3 + 1) * plane;
  float* bp = out + ((size_t)b * 3 + 2) * plane;
  constexpr int NVEC = (WT / 4) * HT;       // 1024 float4 slots per tile
  for (int j = tid; j < NVEC; j += NT) {
    int oy  = j >> 5;                       // j / (WT/4)
    int ox4 = (j & 31) << 2;                // (j % (WT/4)) * 4
    int gy  = gy0 + oy;
    bool yodd = (gy & 1) != 0;

    // Window: green-tile rows oy..oy+2, cols ox4..ox4+5 (pixels are at row oy+1,
    // cols ox4+1..ox4+4). Load once, reuse across the 4 pixels.
    int rb = oy * GTW + ox4;
    float d0[6], d1[6], d2[6], g4[4];
#pragma unroll
    for (int t = 0; t < 6; ++t) {
      d0[t] = s_d[rb           + t];
      d1[t] = s_d[rb +     GTW + t];
      d2[t] = s_d[rb + 2 * GTW + t];
    }
#pragma unroll
    for (int k = 0; k < 4; ++k) g4[k] = s_g[rb + GTW + (k + 1)];

    v4f rr, gg, bb;
#pragma unroll
    for (int k = 0; k < 4; ++k) {           // gx parity == k parity (gx0, ox4 even)
      float c  = d1[k + 1];
      float sh = d1[k] + d1[k + 2];
      float sv = d0[k + 1] + d2[k + 1];
      float sc = (d0[k] + d0[k + 2]) + (d2[k] + d2[k + 2]);
      float cr, cb;
      if ((k & 1) == 0) {                   // gx even
        cr = yodd ? 0.25f * sc : 0.5f * sh;
        cb = yodd ? c          : 0.5f * sv;
      } else {                              // gx odd
        cr = yodd ? 0.5f * sv  : c;
        cb = yodd ? 0.5f * sh  : 0.25f * sc;
      }
      rr[k] = g4[k] + cr;
      gg[k] = g4[k];
      bb[k] = g4[k] + cb;
    }
    size_t o = (size_t)gy * W + (size_t)(gx0 + ox4);
    // Streaming (non-temporal) stores: output is write-once, keep L2 for input.
    __builtin_nontemporal_store(rr, reinterpret_cast<v4f*>(rp + o));
    __builtin_nontemporal_store(gg, reinterpret_cast<v4f*>(gp + o));
    __builtin_nontemporal_store(bb, reinterpret_cast<v4f*>(bp + o));
  }
}

} // namespace

extern "C" void kernel_launch(void* const* d_in, const int* in_sizes, int n_in,
                              void* d_out, int out_size, void* d_ws, size_t ws_size,
                              hipStream_t stream) {
  const float* mosaick = (const float*)d_in[0];
  const float* gfilt   = (const float*)d_in[1];
  const float* gradf   = (const float*)d_in[2];
  float* out = (float*)d_out;
  const int H = 1024, W = 1024;
  const int B = in_sizes[0] / (H * W);      // 32
  dim3 grid(W / WT, H / HT, B);             // 8 x 32 x 32 = 8192 blocks
  demosaic_fused_gfx1250<<<grid, dim3(NT, 1, 1), 0, stream>>>(mosaick, gfilt, gradf, out, H, W);
}